// NegativeCounter_25056839205460
// MI455X (gfx1250) — compile-verified
//
#include <hip/hip_runtime.h>
#include <stdint.h>

// Count-min sketch, D=2 rows, W=2^26 columns, PRIME = 2^31-1 (Mersenne).
// No matrix math in this workload -> WMMA not applicable; this is an
// HBM-random-access bound scatter/gather problem. Optimize the memory path:
// Mersenne-31 reduction (no 64-bit div), f32 global atomics, NT hints on
// streaming traffic, global_prefetch_b8 on gather addresses, wave32 shuffle
// reductions, and the gfx1250 async global<->LDS DMA path (ASYNCcnt) for the
// 512MB table materialization (via inline asm: the clang builtins use
// __device__/__shared__ LangAS params that HIP source cannot spell).

typedef float v4f __attribute__((ext_vector_type(4)));   // native 128-bit vector

static constexpr uint32_t PRIME31 = 2147483647u;      // 2^31 - 1
static constexpr uint32_t WIDTH   = 1u << 26;         // sketch width
static constexpr uint32_t W_MASK  = WIDTH - 1u;

// ((a*id + b) mod (2^31-1)) mod 2^26, exact, branch-light Mersenne fold.
// a,b,id < 2^31 so a*id+b < 2^62.
__device__ __forceinline__ uint32_t cm_hash(uint32_t id, uint32_t a, uint32_t b) {
  uint64_t x = (uint64_t)a * (uint64_t)id + (uint64_t)b;
  x = (x & PRIME31) + (x >> 31);   // < 2^32
  x = (x & PRIME31) + (x >> 31);   // <= 2^31
  uint32_t r = (uint32_t)x;
  if (r >= PRIME31) r -= PRIME31;  // exact residue in [0, PRIME31)
  return r & W_MASK;               // mod 2^26
}

// Kernel 1: materialize working table in d_ws (inputs must not be mutated)
// and zero the total_ct output slot. 128-bit streaming copy routed through
// the gfx1250 async global->LDS->global DMA path: data never transits VGPRs,
// and completion is tracked with ASYNCcnt (s_wait_asynccnt).
__global__ void cm_init(const v4f* __restrict__ src, v4f* __restrict__ dst,
                        uint32_t n4, float* __restrict__ total_slot) {
  uint32_t i      = blockIdx.x * blockDim.x + threadIdx.x;
  uint32_t stride = gridDim.x * blockDim.x;
  if (i == 0) *total_slot = 0.0f;

  __shared__ v4f stage[256];                       // one 16B slot per thread
  // Generic pointer to __shared__: low 32 bits are the LDS byte offset
  // (flat aperture mapping).
  uint32_t lds_off = (uint32_t)(uintptr_t)&stage[threadIdx.x];

  for (; i < n4; i += stride) {
    uint64_t gsrc = (uint64_t)(uintptr_t)(src + i);
    uint64_t gdst = (uint64_t)(uintptr_t)(dst + i);
    // Memory -> LDS (ASYNCcnt++)
    asm volatile("global_load_async_to_lds_b128 %0, %1, off"
                 :
                 : "v"(lds_off), "v"(gsrc)
                 : "memory");
    asm volatile("s_wait_asynccnt 0x0" ::: "memory");   // LDS slot filled
    // LDS -> Memory (ASYNCcnt++)
    asm volatile("global_store_async_from_lds_b128 %0, %1, off"
                 :
                 : "v"(gdst), "v"(lds_off)
                 : "memory");
    asm volatile("s_wait_asynccnt 0x0" ::: "memory");   // slot drained before reuse
  }
}

// Kernel 2: scatter-add one count per id into each sketch row, and stream the
// ids (as float) into the third output segment with non-temporal stores.
// atomicAdd of +1.0f is order-independent here (integer-valued counts < 2^24).
__global__ void cm_update(const int* __restrict__ ids, float* __restrict__ table,
                          const int* __restrict__ ha, const int* __restrict__ hb,
                          float* __restrict__ out_ids, int n) {
  uint32_t a0 = (uint32_t)ha[0], b0 = (uint32_t)hb[0];
  uint32_t a1 = (uint32_t)ha[1], b1 = (uint32_t)hb[1];
  int i = blockIdx.x * blockDim.x + threadIdx.x;
  if (i >= n) return;
  int id_raw = __builtin_nontemporal_load(ids + i);
  uint32_t id = (uint32_t)id_raw;
  uint32_t h0 = cm_hash(id, a0, b0);
  uint32_t h1 = cm_hash(id, a1, b1);
  atomicAdd(table + h0, 1.0f);                     // global_atomic_add_f32
  atomicAdd(table + WIDTH + h1, 1.0f);
  __builtin_nontemporal_store((float)id_raw, out_ids + i);
}

// Kernel 3: gather per-row counts, min across the two rows (count-min
// estimate). Table is 512MB (> 192MB L2) with ~0 reuse per line -> NT loads;
// issue global_prefetch_b8 on both lines before the dependent gathers.
__global__ void cm_query(const int* __restrict__ ids, const float* __restrict__ table,
                         const int* __restrict__ ha, const int* __restrict__ hb,
                         float* __restrict__ out, int n) {
  uint32_t a0 = (uint32_t)ha[0], b0 = (uint32_t)hb[0];
  uint32_t a1 = (uint32_t)ha[1], b1 = (uint32_t)hb[1];
  int i = blockIdx.x * blockDim.x + threadIdx.x;
  if (i >= n) return;
  uint32_t id = (uint32_t)__builtin_nontemporal_load(ids + i);
  const float* p0 = table + cm_hash(id, a0, b0);
  const float* p1 = table + WIDTH + cm_hash(id, a1, b1);
  __builtin_prefetch(p0, 0, 0);                    // global_prefetch_b8
  __builtin_prefetch(p1, 0, 0);
  float c0 = __builtin_nontemporal_load(p0);
  float c1 = __builtin_nontemporal_load(p1);
  __builtin_nontemporal_store(fminf(c0, c1), out + i);
}

// Kernel 4: total_ct = sum of updated row 0. Grid-stride 128-bit NT loads,
// wave32 shuffle reduce, LDS across the 8 waves of the block, one f32 atomic
// per block. All partials are exact integers (total = 2^22 < 2^24) ->
// bit-deterministic regardless of atomic order.
__global__ void cm_total(const v4f* __restrict__ row0, uint32_t n4,
                         float* __restrict__ total) {
  __shared__ float warp_sums[8];                   // 256 threads / wave32
  uint32_t i      = blockIdx.x * blockDim.x + threadIdx.x;
  uint32_t stride = gridDim.x * blockDim.x;
  float s = 0.0f;
  for (; i < n4; i += stride) {
    v4f v = __builtin_nontemporal_load(row0 + i);
    s += (v.x + v.y) + (v.z + v.w);
  }
  // wave32 reduction
  for (int off = 16; off >= 1; off >>= 1) s += __shfl_down(s, off, 32);
  int lane = threadIdx.x & 31;
  int wid  = threadIdx.x >> 5;
  if (lane == 0) warp_sums[wid] = s;
  __syncthreads();
  if (threadIdx.x < 8) {
    s = warp_sums[threadIdx.x];
    for (int off = 4; off >= 1; off >>= 1) s += __shfl_down(s, off, 32);
    if (threadIdx.x == 0) atomicAdd(total, s);
  }
}

extern "C" void kernel_launch(void* const* d_in, const int* in_sizes, int n_in,
                              void* d_out, int out_size, void* d_ws, size_t ws_size,
                              hipStream_t stream) {
  // Input order (setup_inputs): ids[N], table[D*W], hash_a[D], hash_b[D], sync[1]
  const int*   ids      = (const int*)d_in[0];
  const float* table_in = (const float*)d_in[1];
  const int*   hash_a   = (const int*)d_in[2];
  const int*   hash_b   = (const int*)d_in[3];
  (void)d_in[4]; (void)n_in; (void)ws_size; (void)out_size;

  const int      n        = in_sizes[0];            // 1<<22
  const uint32_t tab_elts = (uint32_t)in_sizes[1];  // D*W = 1<<27 floats (512MB)

  float* table      = (float*)d_ws;                 // working copy of the sketch
  float* out_approx = (float*)d_out;                // [N]
  float* out_total  = out_approx + n;               // [1]
  float* out_ids    = out_total + 1;                // [N]

  const int TPB = 256;

  // 1) copy input table -> workspace, zero total slot (async LDS DMA copy)
  {
    uint32_t n4 = tab_elts / 4;                     // 1<<25
    int blocks = 4096;                              // grid-stride
    cm_init<<<blocks, TPB, 0, stream>>>((const v4f*)table_in, (v4f*)table,
                                        n4, out_total);
  }
  // 2) scatter-add counts + stream ids out
  {
    int blocks = (n + TPB - 1) / TPB;
    cm_update<<<blocks, TPB, 0, stream>>>(ids, table, hash_a, hash_b, out_ids, n);
  }
  // 3) gather + min
  {
    int blocks = (n + TPB - 1) / TPB;
    cm_query<<<blocks, TPB, 0, stream>>>(ids, table, hash_a, hash_b, out_approx, n);
  }
  // 4) reduce row 0 -> total_ct
  {
    uint32_t n4 = (tab_elts / 2) / 4;               // WIDTH/4 = 1<<24
    int blocks = 2048;                              // grid-stride
    cm_total<<<blocks, TPB, 0, stream>>>((const v4f*)table, n4, out_total);
  }
}